// LowFrequencyConsistencyLoss_3006477107870
// MI455X (gfx1250) — compile-verified
//
#include <hip/hip_runtime.h>
#include <hip/hip_bf16.h>

typedef __attribute__((ext_vector_type(2))) float v2f;
typedef __attribute__((ext_vector_type(8))) float v8f;

#define NPTS 4096
#define NBATCH 8
#define NTILES (NPTS / 16)    // 256 column tiles per batch
#define WAVES_PER_BLOCK 8
#define TILES_PER_WAVE (NTILES / WAVES_PER_BLOCK)  // 32, compile-time constant
#define ROWBLOCKS (NPTS / 32) // 128 row blocks (32 rows each) per batch
#define BN (NBATCH * NPTS)    // 32768 points total
#define FEAT_BLOCKS (BN / 256)
#define EPS2 1e-12f
#define FLT_BIG 3.4e38f

// Cross-half-wave swap (lanes 0-15 <-> 16-31) without touching the LDS pipe.
__device__ __forceinline__ float lfcl_swap16(float v) {
#if __has_builtin(__builtin_amdgcn_permlanex16)
    int r = __builtin_amdgcn_permlanex16(
        __float_as_int(v), __float_as_int(v),
        0x76543210, 0xfedcba98, /*fi=*/true, /*bound_ctrl=*/false);
    return __int_as_float(r);
#else
    return __shfl_xor(v, 16, 32);
#endif
}

// ---------------------------------------------------------------------------
// Kernel 1: initialize row-min / col-min (squared-distance) buffers to +inf
// ---------------------------------------------------------------------------
__global__ __launch_bounds__(256) void lfcl_init_kernel(unsigned int* mins) {
    int idx = blockIdx.x * blockDim.x + threadIdx.x;
    if (idx < 2 * BN) mins[idx] = 0x7F800000u;  // +inf
}

// ---------------------------------------------------------------------------
// Kernel 2: Chamfer tiles via V_WMMA_F32_16X16X4_F32 on SQUARED distances.
// Block owns 32 rows (two A fragments); each column tile costs one shared B
// fragment + two back-to-back WMMAs. y-loads are software-pipelined one tile
// ahead. sqrt/clamp deferred to the final reduction.
// ---------------------------------------------------------------------------
__global__ __launch_bounds__(256) void lfcl_chamfer_kernel(
    const float* __restrict__ xp,       // [B, N, 3] orig points
    const float* __restrict__ yp,       // [B, N, 3] gen points
    unsigned int* __restrict__ rowmin,  // [B*N] uint-encoded min d^2
    unsigned int* __restrict__ colmin)  // [B*N]
{
    const int blk     = blockIdx.x;
    const int b       = blk >> 7;             // batch  (128 row-blocks/batch)
    const int rowBase = (blk & 127) * 32;
    const int tid     = threadIdx.x;
    const int wave    = tid >> 5;
    const int lane    = tid & 31;
    const int lo      = lane & 15;
    const int hi      = lane >> 4;            // half-wave select

    const float* x = xp + (size_t)b * NPTS * 3;
    const float* y = yp + (size_t)b * NPTS * 3;

    // --- Two A fragments: rows [rowBase, rowBase+16) and [rowBase+16, rowBase+32)
    // 16x4 f32 layout: lanes 0-15 hold K={0,1}, lanes 16-31 hold K={2,3(=0)}.
    const int r0 = rowBase + lo;
    const int r1 = rowBase + 16 + lo;
    const float a00 = x[r0 * 3 + 0], a01 = x[r0 * 3 + 1], a02 = x[r0 * 3 + 2];
    const float a10 = x[r1 * 3 + 0], a11 = x[r1 * 3 + 1], a12 = x[r1 * 3 + 2];
    v2f afrag0, afrag1;
    afrag0.x = hi ? a02 : a00;  afrag0.y = hi ? 0.0f : a01;
    afrag1.x = hi ? a12 : a10;  afrag1.y = hi ? 0.0f : a11;
    const float myx2a = a00 * a00 + a01 * a01 + a02 * a02;
    const float myx2b = a10 * a10 + a11 * a11 + a12 * a12;

    // r2a[i]/r2b[i] = |x_row|^2 for C-fragment row (i + 8*hi) of each group
    float r2a[8], r2b[8];
#pragma unroll
    for (int i = 0; i < 8; ++i) {
        r2a[i] = __shfl(myx2a, i + 8 * hi, 32);
        r2b[i] = __shfl(myx2b, i + 8 * hi, 32);
    }

    float racc[16];
#pragma unroll
    for (int i = 0; i < 16; ++i) racc[i] = FLT_BIG;

    // software-pipelined y loads: fetch tile t's column coords one iter ahead
    int c = wave * 16 + lo;
    float yv0 = y[c * 3 + 0];
    float yv1 = y[c * 3 + 1];
    float yv2 = y[c * 3 + 2];

    for (int t = 0; t < TILES_PER_WAVE; ++t) {   // static trip count: 32
        const int cNext = ((t + 1 < TILES_PER_WAVE)
                               ? (wave + (t + 1) * WAVES_PER_BLOCK) : wave) * 16 + lo;
        const float nv0 = y[cNext * 3 + 0];
        const float nv1 = y[cNext * 3 + 1];
        const float nv2 = y[cNext * 3 + 2];

        // B fragment: 4x16 f32, column = c (same half-wave K split as A)
        v2f bfrag;
        bfrag.x = hi ? yv2 : yv0;
        bfrag.y = hi ? 0.0f : yv1;
        const float c2 = yv0 * yv0 + yv1 * yv1 + yv2 * yv2;

        v8f acc0 = {}, acc1 = {};
        acc0 = __builtin_amdgcn_wmma_f32_16x16x4_f32(
            false, afrag0, false, bfrag, (short)0, acc0, false, false);
        acc1 = __builtin_amdgcn_wmma_f32_16x16x4_f32(
            false, afrag1, false, bfrag, (short)0, acc1, false, false);

        // Epilogue on squared distances: d2 = r2 + c2 - 2*xy
        float colm = FLT_BIG;
#pragma unroll
        for (int i = 0; i < 8; ++i) {
            float d2a = (r2a[i] + c2) - 2.0f * acc0[i];
            float d2b = (r2b[i] + c2) - 2.0f * acc1[i];
            racc[i]     = fminf(racc[i], d2a);
            racc[i + 8] = fminf(racc[i + 8], d2b);
            colm = fminf(colm, fminf(d2a, d2b));
        }
        // column min over all 32 rows: combine the two half-waves, clamp once
        float cm = fminf(colm, lfcl_swap16(colm));
        cm = fmaxf(cm, EPS2);
        if (hi == 0)
            atomicMin(&colmin[(size_t)b * NPTS + c], __float_as_uint(cm));

        c = cNext;
        yv0 = nv0; yv1 = nv1; yv2 = nv2;
    }

    // row min: reduce over the 16 lanes (columns) of each half-wave group
#pragma unroll
    for (int i = 0; i < 16; ++i) {
#pragma unroll
        for (int off = 1; off < 16; off <<= 1)
            racc[i] = fminf(racc[i], __shfl_xor(racc[i], off, 32));
        racc[i] = fmaxf(racc[i], EPS2);
    }
    unsigned int* rmb = rowmin + (size_t)b * NPTS + rowBase;
#pragma unroll
    for (int i = 0; i < 8; ++i) {
        if (lo == i) {
            atomicMin(&rmb[i + 8 * hi],      __float_as_uint(racc[i]));
            atomicMin(&rmb[16 + i + 8 * hi], __float_as_uint(racc[i + 8]));
        }
    }
}

// ---------------------------------------------------------------------------
// Kernel 3: per-point cosine similarity, deterministic block partial sums
// ---------------------------------------------------------------------------
__global__ __launch_bounds__(256) void lfcl_feature_kernel(
    const float* __restrict__ f1, const float* __restrict__ f2,
    float* __restrict__ partial)
{
    const int idx = blockIdx.x * blockDim.x + threadIdx.x;  // point 0..BN-1
    const float* a = f1 + (size_t)idx * 32;
    const float* g = f2 + (size_t)idx * 32;
    float dot = 0.0f, n1 = 0.0f, n2 = 0.0f;
#pragma unroll
    for (int i = 0; i < 32; ++i) {
        float u = a[i], v = g[i];
        dot += u * v;
        n1  += u * u;
        n2  += v * v;
    }
    float cs = dot / (fmaxf(__builtin_amdgcn_sqrtf(n1), 1e-12f) *
                      fmaxf(__builtin_amdgcn_sqrtf(n2), 1e-12f));

    __shared__ float s[256];
    s[threadIdx.x] = cs;
    __syncthreads();
#pragma unroll
    for (int stride = 128; stride > 0; stride >>= 1) {
        if (threadIdx.x < stride) s[threadIdx.x] += s[threadIdx.x + stride];
        __syncthreads();
    }
    if (threadIdx.x == 0) partial[blockIdx.x] = s[0];
}

// ---------------------------------------------------------------------------
// Kernel 4: final reduction -> scalar loss (applies the deferred sqrt here)
// ---------------------------------------------------------------------------
__global__ __launch_bounds__(256) void lfcl_final_kernel(
    const unsigned int* __restrict__ rowmin,
    const unsigned int* __restrict__ colmin,
    const float* __restrict__ partial,
    float* __restrict__ out)
{
    __shared__ float s[256];
    float sr = 0.0f, sc = 0.0f, sp = 0.0f;
    for (int i = threadIdx.x; i < BN; i += 256) {
        sr += __builtin_amdgcn_sqrtf(fmaxf(__uint_as_float(rowmin[i]), EPS2));
        sc += __builtin_amdgcn_sqrtf(fmaxf(__uint_as_float(colmin[i]), EPS2));
    }
    for (int i = threadIdx.x; i < FEAT_BLOCKS; i += 256) sp += partial[i];

    float sums[3] = {sr, sc, sp};
    float res[3];
#pragma unroll
    for (int k = 0; k < 3; ++k) {
        s[threadIdx.x] = sums[k];
        __syncthreads();
        for (int stride = 128; stride > 0; stride >>= 1) {
            if (threadIdx.x < stride) s[threadIdx.x] += s[threadIdx.x + stride];
            __syncthreads();
        }
        res[k] = s[0];
        __syncthreads();
    }

    if (threadIdx.x == 0) {
        const float inv = 1.0f / (float)BN;
        float spatial = res[0] * inv + res[1] * inv;   // mean(min1) + mean(min2)
        float feature = 1.0f - res[2] * inv;           // 1 - mean(cos)
        out[0] = 0.9f * spatial + 0.1f * feature;
    }
}

// ---------------------------------------------------------------------------
extern "C" void kernel_launch(void* const* d_in, const int* in_sizes, int n_in,
                              void* d_out, int out_size, void* d_ws, size_t ws_size,
                              hipStream_t stream) {
    const float* orig_pts = (const float*)d_in[0];   // [8,4096,3]
    const float* gen_pts  = (const float*)d_in[1];   // [8,4096,3]
    const float* orig_ft  = (const float*)d_in[2];   // [8,4096,32]
    const float* gen_ft   = (const float*)d_in[3];   // [8,4096,32]
    float* out = (float*)d_out;

    // workspace layout
    unsigned int* rowmin  = (unsigned int*)d_ws;           // BN uints
    unsigned int* colmin  = rowmin + BN;                   // BN uints
    float*        partial = (float*)(colmin + BN);         // FEAT_BLOCKS floats

    // 1) init min buffers to +inf
    lfcl_init_kernel<<<(2 * BN + 255) / 256, 256, 0, stream>>>(rowmin);

    // 2) Chamfer tiles via WMMA: one block per (batch, 32-row block)
    lfcl_chamfer_kernel<<<NBATCH * ROWBLOCKS, 256, 0, stream>>>(
        orig_pts, gen_pts, rowmin, colmin);

    // 3) feature cosine partial sums
    lfcl_feature_kernel<<<FEAT_BLOCKS, 256, 0, stream>>>(orig_ft, gen_ft, partial);

    // 4) final scalar
    lfcl_final_kernel<<<1, 256, 0, stream>>>(rowmin, colmin, partial, out);
}